// Weighted_least_squares_12713103197136
// MI455X (gfx1250) — compile-verified
//
#include <hip/hip_runtime.h>
#include <hip/hip_bf16.h>

// Problem constants (from reference)
#define N_IMG   32
#define NCH     4
#define HW_TOT  131072
#define SLICES  32
#define HB      (HW_TOT / SLICES)   // 4096 h per block
#define THREADS 256
#define RSTRIDE 36                  // LDS row stride in floats (padded, 16B aligned)
#define REGV    0.001

typedef __attribute__((ext_vector_type(2))) float v2f;
typedef __attribute__((ext_vector_type(8))) float v8f;

// ---------------------------------------------------------------------------
// zero the 32*32 accumulator area in workspace
__global__ void zero_ws(float* ws) {
    int i = blockIdx.x * blockDim.x + threadIdx.x;
    if (i < N_IMG * 32) ws[i] = 0.0f;
}

// ---------------------------------------------------------------------------
// Streaming moments: per (n, h-slice) block, accumulate
//   acc[c*8 + k]   = sum w_c^2 * y^k      (k = 0..4)
//   acc[c*8 + 5+k] = sum w_c^2 * x * y^k  (k = 0..2)
__global__ __launch_bounds__(THREADS)
void moments_kernel(const float* __restrict__ W,
                    const float* __restrict__ grid,
                    float* __restrict__ ws) {
    __shared__ float lds[THREADS * RSTRIDE];

    const int tid   = threadIdx.x;
    const int n     = blockIdx.x / SLICES;
    const int slice = blockIdx.x % SLICES;

    const float* Wn = W    + (size_t)n * NCH * HW_TOT;
    const float* Gn = grid + (size_t)n * HW_TOT * 2;   // interleaved (x, gy)

    float acc[32];
    #pragma unroll
    for (int a = 0; a < 32; ++a) acc[a] = 0.0f;

    #pragma unroll 2
    for (int it = 0; it < HB / (THREADS * 4); ++it) {
        const int h = slice * HB + it * (THREADS * 4) + tid * 4;

        float4 w[NCH];
        #pragma unroll
        for (int c = 0; c < NCH; ++c)
            w[c] = *(const float4*)(Wn + (size_t)c * HW_TOT + h);

        float4 g01 = *(const float4*)(Gn + 2 * h);      // (x0,gy0,x1,gy1)
        float4 g23 = *(const float4*)(Gn + 2 * h + 4);  // (x2,gy2,x3,gy3)

        const float xs[4] = { g01.x, g01.z, g23.x, g23.z };
        const float ys[4] = { 1.0f - g01.y, 1.0f - g01.w,
                              1.0f - g23.y, 1.0f - g23.w };

        #pragma unroll
        for (int s = 0; s < 4; ++s) {
            const float x  = xs[s], y = ys[s];
            const float y2 = y * y;
            const float y3 = y2 * y;
            const float y4 = y2 * y2;
            const float xy = x * y;
            const float xy2 = x * y2;
            #pragma unroll
            for (int c = 0; c < NCH; ++c) {
                const float wv = (s == 0) ? w[c].x : (s == 1) ? w[c].y
                               : (s == 2) ? w[c].z : w[c].w;
                const float w2 = wv * wv;
                acc[c*8+0] += w2;
                acc[c*8+1] = fmaf(w2, y,   acc[c*8+1]);
                acc[c*8+2] = fmaf(w2, y2,  acc[c*8+2]);
                acc[c*8+3] = fmaf(w2, y3,  acc[c*8+3]);
                acc[c*8+4] = fmaf(w2, y4,  acc[c*8+4]);
                acc[c*8+5] = fmaf(w2, x,   acc[c*8+5]);
                acc[c*8+6] = fmaf(w2, xy,  acc[c*8+6]);
                acc[c*8+7] = fmaf(w2, xy2, acc[c*8+7]);
            }
        }
    }

    // spill per-thread partials: row = tid, 32 columns (accumulators)
    #pragma unroll
    for (int a = 0; a < 32; ++a) lds[tid * RSTRIDE + a] = acc[a];
    __syncthreads();

    // fold 256 rows -> 4 rows with vector adds
    for (int s = 128; s >= 4; s >>= 1) {
        if (tid < s) {
            #pragma unroll
            for (int i = 0; i < 8; ++i) {
                float4* dst = (float4*)(lds + tid * RSTRIDE) + i;
                float4  src = *((float4*)(lds + (tid + s) * RSTRIDE) + i);
                dst->x += src.x; dst->y += src.y;
                dst->z += src.z; dst->w += src.w;
            }
        }
        __syncthreads();
    }

    // Final 4-row fold as a WMMA column-sum: D(16x16) = ones(16x4) x B(4x16).
    // f32 16x16x4 layouts: A lane(l%16)=M, v0/v1 hold K={0,1} (lanes 0-15) or
    // K={2,3} (lanes 16-31); B column n = lane(n%16), v0/v1 hold K rows
    // {0,1} / {2,3} by lane half. With A == all-ones, every row of D equals
    // the column sums of B, so D[0] in each lane is the fully reduced value.
    if (tid < 32) {                      // one full wave -> EXEC all 1s
        const int lane = tid;
        const int col  = lane & 15;
        const int kr   = (lane < 16) ? 0 : 2;

        v2f ones = { 1.0f, 1.0f };
        v8f cz   = {};
        v2f bA, bB;
        bA[0] = lds[(kr + 0) * RSTRIDE + col];
        bA[1] = lds[(kr + 1) * RSTRIDE + col];
        bB[0] = lds[(kr + 0) * RSTRIDE + col + 16];
        bB[1] = lds[(kr + 1) * RSTRIDE + col + 16];

        v8f dA = __builtin_amdgcn_wmma_f32_16x16x4_f32(
                     false, ones, false, bA, (short)0, cz, false, false);
        v8f dB = __builtin_amdgcn_wmma_f32_16x16x4_f32(
                     false, ones, false, bB, (short)0, cz, false, false);

        const float total = (lane < 16) ? dA[0] : dB[0];
        atomicAdd(&ws[n * 32 + lane], total);
    }
}

// ---------------------------------------------------------------------------
// 128 independent symmetric 3x3 solves (f64 for conditioning), scatter to
// tuple-concatenated output: out[c*96 + n*3 + p]
__global__ void solve_kernel(const float* __restrict__ ws, float* __restrict__ out) {
    const int i = blockIdx.x * blockDim.x + threadIdx.x;
    if (i >= N_IMG * NCH) return;
    const int n = i >> 2, c = i & 3;
    const float* m = ws + n * 32 + c * 8;

    const double M0 = m[0], M1 = m[1], M2 = m[2], M3 = m[3], M4 = m[4];
    const double B0 = m[5], B1 = m[6], B2 = m[7];

    // Z = [[M4,M3,M2],[M3,M2,M1],[M2,M1,M0]] + REG*I ;  X = [B2,B1,B0]
    const double a00 = M4 + REGV, a01 = M3,        a02 = M2;
    const double a11 = M2 + REGV, a12 = M1;
    const double a22 = M0 + REGV;
    const double x0 = B2, x1 = B1, x2 = B0;

    const double i00 = a11*a22 - a12*a12;
    const double i01 = a02*a12 - a01*a22;
    const double i02 = a01*a12 - a02*a11;
    const double i11 = a00*a22 - a02*a02;
    const double i12 = a02*a01 - a00*a12;
    const double i22 = a00*a11 - a01*a01;
    const double invdet = 1.0 / (a00*i00 + a01*i01 + a02*i02);

    const double b0 = (i00*x0 + i01*x1 + i02*x2) * invdet;
    const double b1 = (i01*x0 + i11*x1 + i12*x2) * invdet;
    const double b2 = (i02*x0 + i12*x1 + i22*x2) * invdet;

    float* o = out + c * (N_IMG * 3) + n * 3;
    o[0] = (float)b0;
    o[1] = (float)b1;
    o[2] = (float)b2;
}

// ---------------------------------------------------------------------------
extern "C" void kernel_launch(void* const* d_in, const int* in_sizes, int n_in,
                              void* d_out, int out_size, void* d_ws, size_t ws_size,
                              hipStream_t stream) {
    const float* W    = (const float*)d_in[0];   // (N*NC, HW) f32
    const float* grid = (const float*)d_in[1];   // (N, HW, 2) f32
    float* out = (float*)d_out;                  // 4 * 32 * 3 f32
    float* ws  = (float*)d_ws;                   // 1024 f32 accumulators

    zero_ws<<<dim3(4), dim3(256), 0, stream>>>(ws);
    moments_kernel<<<dim3(N_IMG * SLICES), dim3(THREADS), 0, stream>>>(W, grid, ws);
    solve_kernel<<<dim3(1), dim3(128), 0, stream>>>(ws, out);
}